// MessageBuildingLayerLSH_19207093748407
// MI455X (gfx1250) — compile-verified
//
#include <hip/hip_runtime.h>
#include <stddef.h>
#include <stdint.h>

typedef float v2f __attribute__((ext_vector_type(2)));
typedef float v8f __attribute__((ext_vector_type(8)));

// Problem constants (from reference)
constexpr int B  = 16;
constexpr int N  = 16384;
constexpr int DM = 128;
constexpr int DN = 256;
constexpr int NB = 128;   // n_bins
constexpr int BS = 128;   // bin_size
constexpr int ROT_STRIDE = 100;  // MAX_NUM_BINS/2

// d_out layout (4-byte elements), reference return order
constexpr size_t OFF_BINS  = 0;
constexpr size_t OFF_XNODE = (size_t)B * NB * BS;                    // 262144
constexpr size_t OFF_DM    = OFF_XNODE + (size_t)B * NB * BS * DN;   // +67108864
constexpr size_t OFF_MSK   = OFF_DM + (size_t)B * NB * BS * BS;      // +33554432

// ---------------------------------------------------------------------------
// K1: LSH projection via V_WMMA_F32_16X16X4_F32 + signed argmax -> bin_idx
// Block: 64 threads (2 waves). Each wave = one 16-row M tile, 4 N tiles (64 cols).
// ---------------------------------------------------------------------------
__global__ __launch_bounds__(64) void k_lsh(const float* __restrict__ xmsg,
                                            const float* __restrict__ rot,
                                            const unsigned char* __restrict__ msk,
                                            int* __restrict__ bin_idx) {
  __shared__ float XS[32 * 132];   // 32 rows x 128 (padded)
  __shared__ float RS[128 * 72];   // 128 x 64 (padded)
  __shared__ float MS[32 * 68];    // mul results for argmax

  const int b = blockIdx.y;
  const int rowbase = blockIdx.x * 32;
  const int tid = threadIdx.x;

  // Load X tile (coalesced float4)
  {
    const float4* src = (const float4*)(xmsg + ((size_t)b * N + rowbase) * DM);
    for (int i = tid; i < 32 * 32; i += 64) {
      const int r = i >> 5, c4 = i & 31;
      float4 v = src[r * 32 + c4];
      float* dst = &XS[r * 132 + c4 * 4];
      dst[0] = v.x; dst[1] = v.y; dst[2] = v.z; dst[3] = v.w;
    }
  }
  // Load rot (128 x 64, global row stride 100)
  for (int i = tid; i < 128 * 64; i += 64) {
    const int d = i >> 6, j = i & 63;
    RS[d * 72 + j] = rot[d * ROT_STRIDE + j];
  }
  __syncthreads();

  const int wv = tid >> 5;
  const int lane = tid & 31;
  const int ln = lane & 15;
  const int hi = lane >> 4;

  v8f acc[4];
#pragma unroll
  for (int c = 0; c < 4; ++c) acc[c] = (v8f){0.f,0.f,0.f,0.f,0.f,0.f,0.f,0.f};

  for (int kb = 0; kb < 128; kb += 4) {
    const int k0 = kb + 2 * hi;
    // A fragment (16x4 f32): lane ln = M, VGPR v -> K = k0+v
    v2f a = *(const v2f*)&XS[(wv * 16 + ln) * 132 + k0];
#pragma unroll
    for (int c = 0; c < 4; ++c) {
      // B fragment (4x16 f32): lane ln = N, VGPR v -> K = k0+v
      v2f bb;
      bb.x = RS[(k0)     * 72 + c * 16 + ln];
      bb.y = RS[(k0 + 1) * 72 + c * 16 + ln];
      acc[c] = __builtin_amdgcn_wmma_f32_16x16x4_f32(false, a, false, bb,
                                                     (short)0, acc[c], false, false);
    }
  }

  // Spill mul to LDS (C layout: VGPR j -> row j + 8*hi, lane -> col)
#pragma unroll
  for (int c = 0; c < 4; ++c)
#pragma unroll
    for (int j = 0; j < 8; ++j) {
      const int r = wv * 16 + j + 8 * hi;
      MS[r * 68 + c * 16 + ln] = acc[c][j];
    }
  __syncthreads();

  if (tid < 32) {
    const int r = tid;
    float maxv = MS[r * 68]; int maxj = 0;
    float minv = maxv;       int minj = 0;
    for (int j = 1; j < 64; ++j) {
      const float v = MS[r * 68 + j];
      if (v > maxv) { maxv = v; maxj = j; }
      if (v < minv) { minv = v; minj = j; }
    }
    // argmax over concat([mul, -mul]); ties -> first half (<=)
    const int bin = (maxv >= -minv) ? maxj : (64 + minj);
    const int n = rowbase + r;
    const unsigned char m = msk[(size_t)b * N + n];
    bin_idx[(size_t)b * N + n] = bin + (m ? 0 : (NB - 1));
  }
}

// ---------------------------------------------------------------------------
// K2: stable counting sort per batch (256 possible bin values).
// 32 threads; thread t owns chunk [t*512, t*512+512) -> stability by chunk order.
// ---------------------------------------------------------------------------
__global__ __launch_bounds__(32) void k_sort(const int* __restrict__ bin_idx,
                                             int* __restrict__ bins_out) {
  __shared__ int hist[256 * 32];
  __shared__ int tot[256];
  __shared__ int base[256];
  const int b = blockIdx.x;
  const int t = threadIdx.x;

  for (int i = t; i < 256 * 32; i += 32) hist[i] = 0;
  __syncthreads();

  const int* bi = bin_idx + (size_t)b * N + t * 512;
  for (int e = 0; e < 512; ++e) hist[bi[e] * 32 + t]++;
  __syncthreads();

  // per-bin exclusive scan over thread columns + bin totals
  for (int bb = t * 8; bb < t * 8 + 8; ++bb) {
    int run = 0;
    for (int c = 0; c < 32; ++c) {
      const int v = hist[bb * 32 + c];
      hist[bb * 32 + c] = run;
      run += v;
    }
    tot[bb] = run;
  }
  __syncthreads();
  if (t == 0) {
    int run = 0;
    for (int bb = 0; bb < 256; ++bb) { base[bb] = run; run += tot[bb]; }
  }
  __syncthreads();

  int* bo = bins_out + (size_t)b * N;
  for (int e = 0; e < 512; ++e) {
    const int bv = bi[e];
    const int pos = base[bv] + hist[bv * 32 + t]++;
    bo[pos] = t * 512 + e;
  }
}

// ---------------------------------------------------------------------------
// K3: gather x_node rows (1 KB rows, float4 coalesced) + msk_b
// ---------------------------------------------------------------------------
__global__ __launch_bounds__(64) void k_gather(const float* __restrict__ xnode,
                                               const unsigned char* __restrict__ msk,
                                               const int* __restrict__ bins,
                                               float* __restrict__ xnode_out,
                                               float* __restrict__ msk_out) {
  const int row = blockIdx.x;          // b*16384 + bin*128 + p
  const int b = row >> 14;
  const int src = bins[row];
  const float4* in = (const float4*)(xnode + ((size_t)b * N + src) * DN);
  float4* out = (float4*)(xnode_out + (size_t)row * DN);
  out[threadIdx.x] = in[threadIdx.x];
  if (threadIdx.x == 0)
    msk_out[row] = msk[(size_t)b * N + src] ? 1.0f : 0.0f;
}

// ---------------------------------------------------------------------------
// K4: per-bin 128x128 Gram via V_WMMA_F32_16X16X4_F32 + Gaussian kernel epilogue.
// Tile gather goes straight to LDS with GLOBAL_LOAD_ASYNC_TO_LDS_B128 (ASYNCcnt).
// Masking is applied only in the epilogue (entries touching masked rows are
// zeroed by m_p*m_q, so pre-masking X is unnecessary).
// 256 threads = 8 waves; wave w computes 16x128 strip (8 output tiles).
// ---------------------------------------------------------------------------
__global__ __launch_bounds__(256) void k_kernelmat(const float* __restrict__ xmsg,
                                                   const int* __restrict__ bins,
                                                   const float* __restrict__ mskb,
                                                   float* __restrict__ dm) {
  __shared__ float XS[128 * 132];   // 128 rows x 128, padded stride -> conflict-free
  __shared__ float sq[128];
  __shared__ float ms[128];

  const int blk = blockIdx.x;       // b*128 + bin
  const int b = blk >> 7;
  const int tid = threadIdx.x;

  // Async-gather 128 rows (2 threads per row, 16B granules) directly into LDS.
  {
    const int r = tid >> 1;
    const int h = tid & 1;
    const int src = bins[(size_t)blk * BS + r];
    const float* gp = xmsg + ((size_t)b * N + src) * DM + h * 64;
    const unsigned lo = (unsigned)(uintptr_t)(&XS[r * 132 + h * 64]);
#pragma unroll
    for (int i = 0; i < 16; ++i) {
      asm volatile("global_load_async_to_lds_b128 %0, %1, off"
                   :: "v"(lo + i * 16), "v"(gp + i * 4)
                   : "memory");
    }
  }
  asm volatile("s_wait_asynccnt 0x0" ::: "memory");
  __syncthreads();

  if (tid < 128) {
    float s = 0.f;
    const float* xr = &XS[tid * 132];
    for (int k = 0; k < 128; ++k) s += xr[k] * xr[k];
    sq[tid] = s;
    ms[tid] = mskb[(size_t)blk * BS + tid];
  }
  __syncthreads();

  const int wv = tid >> 5;
  const int lane = tid & 31;
  const int ln = lane & 15;
  const int hi = lane >> 4;

  v8f acc[8];
#pragma unroll
  for (int c = 0; c < 8; ++c) acc[c] = (v8f){0.f,0.f,0.f,0.f,0.f,0.f,0.f,0.f};

  for (int kb = 0; kb < 128; kb += 4) {
    const int k0 = kb + 2 * hi;
    v2f frag[8];
#pragma unroll
    for (int t2 = 0; t2 < 8; ++t2)
      frag[t2] = *(const v2f*)&XS[(t2 * 16 + ln) * 132 + k0];  // ds_load_b64
    const v2f a = frag[wv];
#pragma unroll
    for (int c = 0; c < 8; ++c)
      acc[c] = __builtin_amdgcn_wmma_f32_16x16x4_f32(false, a, false, frag[c],
                                                     (short)0, acc[c], false, false);
  }

  float* out = dm + (size_t)blk * BS * BS;
#pragma unroll
  for (int c = 0; c < 8; ++c) {
#pragma unroll
    for (int j = 0; j < 8; ++j) {
      const int rr = wv * 16 + j + 8 * hi;
      const int q = c * 16 + ln;
      const float g = acc[c][j];
      float d2 = sq[rr] - 2.0f * g + sq[q];
      float D = sqrtf(fmaxf(d2, 1e-6f));
      float v = __expf(-0.1f * D);
      v = fminf(v, 1.0f) * ms[rr] * ms[q];
      out[(size_t)rr * BS + q] = v;
    }
  }
}

// ---------------------------------------------------------------------------
extern "C" void kernel_launch(void* const* d_in, const int* in_sizes, int n_in,
                              void* d_out, int out_size, void* d_ws, size_t ws_size,
                              hipStream_t stream) {
  const float* x_msg = (const float*)d_in[0];
  const float* x_node = (const float*)d_in[1];
  const unsigned char* msk = (const unsigned char*)d_in[2];  // jax bool = 1 byte
  const float* rot = (const float*)d_in[3];

  float* out = (float*)d_out;
  int*   bins      = (int*)(out + OFF_BINS);
  float* xnode_out = out + OFF_XNODE;
  float* dmat      = out + OFF_DM;
  float* mskb      = out + OFF_MSK;

  int* bin_idx = (int*)d_ws;  // B*N ints = 1 MiB scratch

  dim3 g1(N / 32, B);
  k_lsh<<<g1, 64, 0, stream>>>(x_msg, rot, msk, bin_idx);
  k_sort<<<B, 32, 0, stream>>>(bin_idx, bins);
  k_gather<<<B * N, 64, 0, stream>>>(x_node, msk, bins, xnode_out, mskb);
  k_kernelmat<<<B * NB, 256, 0, stream>>>(x_msg, bins, mskb, dmat);
}